// Seq2SeqAttentionMechanism_21638045237354
// MI455X (gfx1250) — compile-verified
//
#include <hip/hip_runtime.h>
#include <math.h>

// Seq2Seq attention, MI455X (gfx1250), wave32.
//
// Shapes: dec (128,128) f32, enc (128,4096,128) f32.
//   score[b,s] = dot(dec[b,:], enc[b,s,:]);  attn = softmax(score); ctx = attn @ enc
// Memory-bound: enc = 256 MB, FLOPs ~268 M -> AI ~1 FLOP/B. Optimal = stream enc
// ONCE (online softmax) ~= 256 MB / 23.3 TB/s ~= 11 us floor.
//
// Pass 1: per-wave flash-style online softmax over a 256-seq span; scores via
//         v_wmma_f32_16x16x4_f32 (dec broadcast in A, enc tile in B -> each
//         lane L holds score[s0+(L&15)] replicated); ctx accumulated with VALU
//         FMAs (lane owns 4 h columns). Raw scores -> attn region of d_out,
//         per-wave (m,l,ctx[128]) partials -> workspace.
// Pass 2: combine 16 partials per batch -> context output + (M_b, 1/total_b).
// Pass 3: attn[i] = exp(raw - M_b) * inv_b  (in-place, 4 MB).

typedef __attribute__((ext_vector_type(2))) float v2f;
typedef __attribute__((ext_vector_type(4))) float v4f;
typedef __attribute__((ext_vector_type(8))) float v8f;

#define BS   128
#define SEQ  4096
#define HID  128
#define WAVES_PER_BATCH 16
#define SEQ_PER_WAVE    (SEQ / WAVES_PER_BATCH)     // 256
#define TILES_PER_WAVE  (SEQ_PER_WAVE / 16)         // 16
#define BLOCK_THREADS   256
#define WAVES_PER_BLOCK (BLOCK_THREADS / 32)        // 8
#define TOTAL_WAVES     (BS * WAVES_PER_BATCH)      // 2048
#define GRID_A          (TOTAL_WAVES / WAVES_PER_BLOCK)
#define PARTIAL_STRIDE  130                         // m, l, ctx[128]

__global__ __launch_bounds__(BLOCK_THREADS)
void attn_pass1(const float* __restrict__ dec, const float* __restrict__ enc,
                float* __restrict__ attn_raw, float* __restrict__ partials)
{
    const int lane = threadIdx.x & 31;
    const int half = lane >> 4;                       // 0: K=0,1  1: K=2,3
    const int wid  = blockIdx.x * WAVES_PER_BLOCK + (threadIdx.x >> 5);
    const int b    = wid / WAVES_PER_BATCH;
    const int seq0 = (wid % WAVES_PER_BATCH) * SEQ_PER_WAVE;

    const float* decb = dec + b * HID;
    const float* encb = enc + (size_t)b * SEQ * HID;

    // A-matrix fragments: A[m,k] = dec[k0+k] for all m (broadcast over rows).
    // Per ISA 16x4 f32 A layout: lane L holds (K=0,1) for L<16, (K=2,3) for L>=16.
    v2f a[32];
    #pragma unroll
    for (int j = 0; j < 32; ++j)
        a[j] = *(const v2f*)(decb + 4 * j + 2 * half);

    float m_run = -__builtin_inff();
    float l_run = 0.0f;
    float cx0 = 0.f, cx1 = 0.f, cx2 = 0.f, cx3 = 0.f;

    const int rowoff = lane & 15;                     // N index: seq row in tile

    for (int t = 0; t < TILES_PER_WAVE; ++t) {
        const int s0 = seq0 + t * 16;
        // B[k,n] = enc[s0+n, k0+k]; lane L supplies row s0+(L&15),
        // K-pair (0,1) for L<16 and (2,3) for L>=16 (layout by C/D analogy).
        const float* bp = encb + (size_t)(s0 + rowoff) * HID + 2 * half;

        v8f c = {0.f, 0.f, 0.f, 0.f, 0.f, 0.f, 0.f, 0.f};
        #pragma unroll
        for (int j = 0; j < 32; ++j) {
            v2f bf = *(const v2f*)(bp + 4 * j);
            c = __builtin_amdgcn_wmma_f32_16x16x4_f32(
                    false, a[j], false, bf, (short)0, c, false, false);
        }
        // D[m,n] = score[s0+n] for every m -> lane L holds score[s0+(L&15)].
        float sc = c[0];

        if (lane < 16)
            attn_raw[(size_t)b * SEQ + s0 + lane] = sc;

        // Tile max / exp-sum across the 16-lane group (halves are duplicates).
        float tmax = sc;
        tmax = fmaxf(tmax, __shfl_xor(tmax, 1, 32));
        tmax = fmaxf(tmax, __shfl_xor(tmax, 2, 32));
        tmax = fmaxf(tmax, __shfl_xor(tmax, 4, 32));
        tmax = fmaxf(tmax, __shfl_xor(tmax, 8, 32));

        const float m_new = fmaxf(m_run, tmax);
        const float corr  = __expf(m_run - m_new);    // first tile: exp(-inf)=0
        const float e     = __expf(sc - m_new);

        float esum = e;
        esum += __shfl_xor(esum, 1, 32);
        esum += __shfl_xor(esum, 2, 32);
        esum += __shfl_xor(esum, 4, 32);
        esum += __shfl_xor(esum, 8, 32);

        l_run = l_run * corr + esum;
        cx0 *= corr; cx1 *= corr; cx2 *= corr; cx3 *= corr;

        // Context accumulation: lane owns h = 4*lane .. 4*lane+3.
        const float* cp = encb + (size_t)s0 * HID + 4 * lane;
        if (t + 1 < TILES_PER_WAVE)
            __builtin_prefetch(cp + 16 * HID, 0, 1); // global_prefetch_b8
        #pragma unroll
        for (int s = 0; s < 16; ++s) {
            const float es = __shfl(e, s, 32);        // broadcast e[s0+s]
            v4f v = *(const v4f*)(cp + s * HID);
            cx0 += es * v.x; cx1 += es * v.y; cx2 += es * v.z; cx3 += es * v.w;
        }
        m_run = m_new;
    }

    float* p = partials + (size_t)wid * PARTIAL_STRIDE;
    if (lane == 0) { p[0] = m_run; p[1] = l_run; }
    v4f cv = { cx0, cx1, cx2, cx3 };
    *(v4f*)(p + 2 + 4 * lane) = cv;
}

__global__ __launch_bounds__(HID)
void attn_pass2(const float* __restrict__ partials, float* __restrict__ ctx_out,
                float* __restrict__ stats)
{
    const int b = blockIdx.x;
    const int h = threadIdx.x;
    const float* p = partials + (size_t)b * WAVES_PER_BATCH * PARTIAL_STRIDE;

    float M = -__builtin_inff();
    #pragma unroll
    for (int w = 0; w < WAVES_PER_BATCH; ++w)
        M = fmaxf(M, p[w * PARTIAL_STRIDE]);

    float total = 0.0f, acc = 0.0f;
    #pragma unroll
    for (int w = 0; w < WAVES_PER_BATCH; ++w) {
        const float* q = p + w * PARTIAL_STRIDE;
        const float scale = __expf(q[0] - M);
        total += scale * q[1];
        acc   += scale * q[2 + h];
    }
    const float inv = 1.0f / total;
    ctx_out[b * HID + h] = acc * inv;
    if (h == 0) { stats[2 * b] = M; stats[2 * b + 1] = inv; }
}

__global__ __launch_bounds__(256)
void attn_pass3(float* __restrict__ attn, const float* __restrict__ stats)
{
    const int i = blockIdx.x * 256 + threadIdx.x;     // grid sized exactly
    const int b = i >> 12;                            // i / 4096
    const float M   = stats[2 * b];
    const float inv = stats[2 * b + 1];
    attn[i] = __expf(attn[i] - M) * inv;
}

extern "C" void kernel_launch(void* const* d_in, const int* in_sizes, int n_in,
                              void* d_out, int out_size, void* d_ws, size_t ws_size,
                              hipStream_t stream)
{
    (void)in_sizes; (void)n_in; (void)out_size; (void)ws_size;
    const float* dec = (const float*)d_in[0];          // (128,128)
    const float* enc = (const float*)d_in[1];          // (128,4096,128)
    float* attn = (float*)d_out;                       // (128,4096)
    float* ctx  = (float*)d_out + (size_t)BS * SEQ;    // (128,128)
    float* partials = (float*)d_ws;                    // 2048*130 floats ~1.04 MB
    float* stats    = partials + (size_t)TOTAL_WAVES * PARTIAL_STRIDE; // 256 floats

    attn_pass1<<<GRID_A, BLOCK_THREADS, 0, stream>>>(dec, enc, attn, partials);
    attn_pass2<<<BS, HID, 0, stream>>>(partials, ctx, stats);
    attn_pass3<<<(BS * SEQ) / 256, 256, 0, stream>>>(attn, stats);
}